// MonotonicityLoss_48455821033938
// MI455X (gfx1250) — compile-verified
//
#include <hip/hip_runtime.h>
#include <hip/hip_bf16.h>

#define MARGIN_F 0.1f
#define BLOCK 256
#define TILE_J 1024

#if defined(__gfx1250__) && __has_builtin(__builtin_amdgcn_global_load_async_to_lds_b128) && __has_builtin(__builtin_amdgcn_s_wait_asynccnt)
#define USE_ASYNC_LDS 1
#else
#define USE_ASYNC_LDS 0
#endif

// ---------------------------------------------------------------------------
// Kernel 1: pack pred + (group<<8 | galloyl) into one 8-byte record.
// With this key encoding, the pair mask (same group AND gal_j < gal_i) is a
// single unsigned range check: (key_j - (group_i<<8)) < gal_i.
// ---------------------------------------------------------------------------
__global__ void mono_pack_kernel(const float* __restrict__ pred,
                                 const int* __restrict__ gal,
                                 const int* __restrict__ grp,
                                 uint2* __restrict__ packed, int n) {
    int idx = blockIdx.x * blockDim.x + threadIdx.x;
    if (idx < n) {
        uint2 v;
        v.x = __float_as_uint(pred[idx]);
        v.y = (((unsigned)grp[idx]) << 8) | ((unsigned)gal[idx] & 0xFFu);
        packed[idx] = v;
    }
}

// ---------------------------------------------------------------------------
// Kernel 2: block = 256 i's (blockIdx.x) x one j-tile of 1024 (blockIdx.y).
// j-tile staged into LDS via CDNA5 async global->LDS copies (ASYNCcnt +
// s_wait_asynccnt); inner loop reads 2 records per ds_load_b128 (broadcast)
// and evaluates each pair with ~6 VALU ops. Deterministic reductions.
// ---------------------------------------------------------------------------
__global__ void __launch_bounds__(BLOCK)
mono_pair_kernel(const uint2* __restrict__ packed,
                 float* __restrict__ psum, unsigned* __restrict__ pcnt, int n) {
    __shared__ alignas(16) uint2 tile[TILE_J];
    __shared__ float wsum[BLOCK / 32];
    __shared__ unsigned wcnt[BLOCK / 32];

    const int tid = threadIdx.x;
    const int jbase = blockIdx.y * TILE_J;

#if USE_ASYNC_LDS
    {
        // Builtin takes non-const pointers to 16-byte int vectors:
        // src in addrspace(1), dst in addrspace(3).
        typedef int v4i __attribute__((vector_size(16)));
        typedef __attribute__((address_space(1))) v4i* gvp_t;
        typedef __attribute__((address_space(3))) v4i* lvp_t;

        uint2* src_nc = const_cast<uint2*>(packed + jbase);
        gvp_t gsrc = (gvp_t)(void*)src_nc;
        lvp_t ldst = (lvp_t)(void*)&tile[0];

        const int nvec = (TILE_J * (int)sizeof(uint2)) / 16;   // 512
        for (int k = tid; k < nvec; k += BLOCK) {
            __builtin_amdgcn_global_load_async_to_lds_b128(
                gsrc + k, ldst + k, /*imm offset*/ 0, /*cpol*/ 0);
        }
        __builtin_amdgcn_s_wait_asynccnt(0);
    }
    __syncthreads();
#else
    for (int k = tid; k < TILE_J; k += BLOCK) tile[k] = packed[jbase + k];
    __syncthreads();
#endif

    // This thread's private i-row.
    const int i = blockIdx.x * BLOCK + tid;
    uint2 me = packed[i];                        // n is a multiple of BLOCK
    const float pi = __uint_as_float(me.x) + MARGIN_F;
    const unsigned base_i = me.y & 0xFFFFFF00u;  // group_i << 8
    const unsigned ai = me.y & 0xFFu;            // gal_i (< 256)

    float sum = 0.0f;
    unsigned cnt = 0u;

    const uint4* tile4 = reinterpret_cast<const uint4*>(&tile[0]);

#pragma unroll 8
    for (int q = 0; q < TILE_J / 2; ++q) {
        const uint4 v = tile4[q];                // ds_load_b128: 2 j-records
        // --- pair 0 ---
        {
            const bool m = (v.y - base_i) < ai;  // same group && gal_j < gal_i
            float d = fmaxf(pi - __uint_as_float(v.x), 0.0f);
            const float r = m ? d : 0.0f;
            sum = fmaf(r, r, sum);
            cnt += (unsigned)m;
        }
        // --- pair 1 ---
        {
            const bool m = (v.w - base_i) < ai;
            float d = fmaxf(pi - __uint_as_float(v.z), 0.0f);
            const float r = m ? d : 0.0f;
            sum = fmaf(r, r, sum);
            cnt += (unsigned)m;
        }
    }

    // Wave32 reduction (fixed xor pattern -> deterministic).
    for (int o = 16; o > 0; o >>= 1) {
        sum += __shfl_xor(sum, o, 32);
        cnt += __shfl_xor(cnt, o, 32);
    }
    if ((tid & 31) == 0) {
        wsum[tid >> 5] = sum;
        wcnt[tid >> 5] = cnt;
    }
    __syncthreads();
    if (tid == 0) {
        float bs = 0.0f;
        unsigned bc = 0u;
        for (int w = 0; w < BLOCK / 32; ++w) { bs += wsum[w]; bc += wcnt[w]; }
        const int part = blockIdx.y * gridDim.x + blockIdx.x;
        psum[part] = bs;
        pcnt[part] = bc;
    }
}

// ---------------------------------------------------------------------------
// Kernel 3: single-block deterministic tree reduction of the partials.
// ---------------------------------------------------------------------------
__global__ void __launch_bounds__(BLOCK)
mono_finalize_kernel(const float* __restrict__ psum,
                     const unsigned* __restrict__ pcnt,
                     float* __restrict__ out, int nparts) {
    __shared__ float fs[BLOCK];
    __shared__ unsigned fc[BLOCK];
    const int t = threadIdx.x;
    float s = 0.0f;
    unsigned c = 0u;
    for (int k = t; k < nparts; k += BLOCK) { s += psum[k]; c += pcnt[k]; }
    fs[t] = s;
    fc[t] = c;
    __syncthreads();
    for (int stride = BLOCK / 2; stride > 0; stride >>= 1) {
        if (t < stride) { fs[t] += fs[t + stride]; fc[t] += fc[t + stride]; }
        __syncthreads();
    }
    if (t == 0)
        out[0] = (fc[0] > 0u) ? (fs[0] / (float)fc[0]) : 0.0f;
}

// ---------------------------------------------------------------------------
extern "C" void kernel_launch(void* const* d_in, const int* in_sizes, int n_in,
                              void* d_out, int out_size, void* d_ws, size_t ws_size,
                              hipStream_t stream) {
    const float* pred = (const float*)d_in[0];
    const int* gal = (const int*)d_in[1];
    const int* grp = (const int*)d_in[2];
    const int n = in_sizes[0];                   // 8192

    // Workspace layout: packed records | per-block sums | per-block counts
    uint2* packed = (uint2*)d_ws;
    const size_t packed_bytes = (size_t)n * sizeof(uint2);
    float* psum = (float*)((char*)d_ws + packed_bytes);
    const int grid_i = n / BLOCK;                // 32
    const int grid_j = n / TILE_J;               // 8
    const int nparts = grid_i * grid_j;          // 256
    unsigned* pcnt = (unsigned*)((char*)psum + (size_t)nparts * sizeof(float));

    mono_pack_kernel<<<(n + BLOCK - 1) / BLOCK, BLOCK, 0, stream>>>(pred, gal, grp, packed, n);

    dim3 grid(grid_i, grid_j);
    mono_pair_kernel<<<grid, BLOCK, 0, stream>>>(packed, psum, pcnt, n);

    mono_finalize_kernel<<<1, BLOCK, 0, stream>>>(psum, pcnt, (float*)d_out, nparts);
}